// GenODE_44641890074925
// MI455X (gfx1250) — compile-verified
//
#include <hip/hip_runtime.h>
#include <math.h>

typedef __attribute__((ext_vector_type(2))) float v2f;
typedef __attribute__((ext_vector_type(8))) float v8f;

// D = A(16x4,f32) * B(4x16,f32) + C(16x16,f32)
#define WMMA4(a, b, c) \
    __builtin_amdgcn_wmma_f32_16x16x4_f32(false, (a), false, (b), (short)0, (c), false, false)

// Branchless tanh: prefer gfx1250's hardware V_TANH_F32; otherwise a clamp +
// v_exp_f32 + v_rcp_f32 sequence with no EXEC-divergent paths.
__device__ __forceinline__ float fast_tanh(float x) {
#if __has_builtin(__builtin_amdgcn_tanhf)
    return __builtin_amdgcn_tanhf(x);
#else
    float xc = fminf(fmaxf(x, -10.0f), 10.0f);
    // exp(2x) = exp2(x * 2*log2(e))
    float t = __builtin_amdgcn_exp2f(xc * 2.8853900817779268f);
    return (t - 1.0f) * __builtin_amdgcn_rcpf(t + 1.0f);
#endif
}

// One wave (32 lanes) integrates 16 consecutive trajectory rows.
// Row (16*g + m) is frozen after exactly (16*g + m) RK45 steps.
__launch_bounds__(32)
__global__ void genode_rk45_wmma(const float* __restrict__ z_end,
                                 const float* __restrict__ W1,  // [16][128]
                                 const float* __restrict__ b1,  // [128]
                                 const float* __restrict__ W2,  // [128][16]
                                 const float* __restrict__ b2,  // [16]
                                 float* __restrict__ out,       // [N][16]
                                 float h)
{
    // Weights pre-swizzled into WMMA B-operand layout (lane = 16*(k-half)+n, vgpr = k%2)
    __shared__ v2f w1B[8][4][32];            // [n-tile][k-chunk][lane]
    __shared__ v2f w2B[32][32];              // [k-chunk][lane]
    __shared__ __align__(16) float hbuf[16][132];  // tanh activations, row-major (padded)
    __shared__ __align__(16) float zbuf[16][20];   // stage input Z', row-major (padded)

    const int lane = threadIdx.x;   // 0..31
    const int n    = lane & 15;     // column within 16-wide tile
    const int half = lane >> 4;     // 0 or 1
    const int dlt  = half * 2;      // K offset inside a 4-wide chunk (A/B layout)
    const int g    = blockIdx.x;    // tile of rows 16g .. 16g+15

    // ---- one-time: weights -> LDS in B layout ----
    #pragma unroll
    for (int t = 0; t < 8; ++t)
        #pragma unroll
        for (int c = 0; c < 4; ++c) {
            int k = 4 * c + dlt;
            v2f v;
            v.x = W1[k * 128 + 16 * t + n];
            v.y = W1[(k + 1) * 128 + 16 * t + n];
            w1B[t][c][lane] = v;
        }
    #pragma unroll
    for (int c = 0; c < 32; ++c) {
        int k = 4 * c + dlt;
        v2f v;
        v.x = W2[k * 16 + n];
        v.y = W2[(k + 1) * 16 + n];
        w2B[c][lane] = v;
    }
    float b1v[8];
    #pragma unroll
    for (int t = 0; t < 8; ++t) b1v[t] = b1[16 * t + n];
    const float b2v = b2[n];

    // ---- state z in WMMA C/D layout: vgpr v holds row m = v + 8*half, column n ----
    v8f z;
    #pragma unroll
    for (int v = 0; v < 8; ++v)
        z[v] = z_end[(16 * g + v + half * 8) * 16 + n];

    __syncthreads();

    // Write a stage input (C/D layout regs) to zbuf row-major.
    auto put_z = [&](const v8f& val) {
        #pragma unroll
        for (int v = 0; v < 8; ++v)
            zbuf[v + half * 8][n] = val[v];
    };

    // f(Z') = tanh(Z' @ W1 + b1) @ W2 + b2 ; Z' read from zbuf; result in C/D layout.
    auto feval = [&]() -> v8f {
        __syncthreads();  // zbuf stores -> A-operand loads
        v2f az[4];
        #pragma unroll
        for (int c = 0; c < 4; ++c)
            az[c] = *(const v2f*)&zbuf[n][4 * c + dlt];  // A: m = n(lane), k = 4c+dlt..+1

        #pragma unroll
        for (int t = 0; t < 8; ++t) {
            v8f acc;
            #pragma unroll
            for (int v = 0; v < 8; ++v) acc[v] = b1v[t];
            #pragma unroll
            for (int c = 0; c < 4; ++c)
                acc = WMMA4(az[c], w1B[t][c][lane], acc);
            #pragma unroll
            for (int v = 0; v < 8; ++v)
                hbuf[v + half * 8][16 * t + n] = fast_tanh(acc[v]);
        }
        __syncthreads();  // hbuf stores -> A-operand loads

        v8f o;
        #pragma unroll
        for (int v = 0; v < 8; ++v) o[v] = b2v;
        #pragma unroll
        for (int c = 0; c < 32; ++c) {
            v2f ah = *(const v2f*)&hbuf[n][4 * c + dlt];
            o = WMMA4(ah, w2B[c][lane], o);
        }
        return o;
    };

    const int nsteps = 16 * g + 15;
    for (int s = 1; s <= nsteps; ++s) {
        put_z(z);
        v8f k1 = feval();

        v8f t;
        #pragma unroll
        for (int v = 0; v < 8; ++v) t[v] = z[v] + h * (0.2f * k1[v]);
        put_z(t);
        v8f k2 = feval();

        #pragma unroll
        for (int v = 0; v < 8; ++v)
            t[v] = z[v] + h * ((3.0f / 40.0f) * k1[v] + (9.0f / 40.0f) * k2[v]);
        put_z(t);
        v8f k3 = feval();

        #pragma unroll
        for (int v = 0; v < 8; ++v)
            t[v] = z[v] + h * ((44.0f / 45.0f) * k1[v] - (56.0f / 15.0f) * k2[v]
                               + (32.0f / 9.0f) * k3[v]);
        put_z(t);
        v8f k4 = feval();

        #pragma unroll
        for (int v = 0; v < 8; ++v)
            t[v] = z[v] + h * ((19372.0f / 6561.0f) * k1[v] - (25360.0f / 2187.0f) * k2[v]
                               + (64448.0f / 6561.0f) * k3[v] - (212.0f / 729.0f) * k4[v]);
        put_z(t);
        v8f k5 = feval();

        #pragma unroll
        for (int v = 0; v < 8; ++v)
            t[v] = z[v] + h * ((9017.0f / 3168.0f) * k1[v] - (355.0f / 33.0f) * k2[v]
                               + (46732.0f / 5247.0f) * k3[v] + (49.0f / 176.0f) * k4[v]
                               - (5103.0f / 18656.0f) * k5[v]);
        put_z(t);
        v8f k6 = feval();

        // 5th-order combination; freeze rows whose record time has passed.
        #pragma unroll
        for (int v = 0; v < 8; ++v) {
            float zn = z[v] + h * ((35.0f / 384.0f) * k1[v] + (500.0f / 1113.0f) * k3[v]
                                   + (125.0f / 192.0f) * k4[v] - (2187.0f / 6784.0f) * k5[v]
                                   + (11.0f / 84.0f) * k6[v]);
            int target = 16 * g + v + half * 8;   // steps this row needs
            z[v] = (target >= s) ? zn : z[v];     // cndmask, EXEC stays all-ones
        }
    }

    // ---- write frozen states: row m = v + 8*half, col n ----
    #pragma unroll
    for (int v = 0; v < 8; ++v)
        out[(16 * g + v + half * 8) * 16 + n] = z[v];
}

extern "C" void kernel_launch(void* const* d_in, const int* in_sizes, int n_in,
                              void* d_out, int out_size, void* d_ws, size_t ws_size,
                              hipStream_t stream) {
    const float* z_end = (const float*)d_in[0];
    const float* W1    = (const float*)d_in[1];
    const float* b1    = (const float*)d_in[2];
    const float* W2    = (const float*)d_in[3];
    const float* b2    = (const float*)d_in[4];
    float* out = (float*)d_out;

    const int N = in_sizes[0] / 16;           // 512 rows
    const float h = 1.0f / (float)(N - 1);    // 1/511
    const int ntiles = N / 16;                // 32 waves, one per 16-row tile

    hipLaunchKernelGGL(genode_rk45_wmma, dim3(ntiles), dim3(32), 0, stream,
                       z_end, W1, b1, W2, b2, out, h);
}